// Model_69776038691103
// MI455X (gfx1250) — compile-verified
//
#include <hip/hip_runtime.h>

// ---------------------------------------------------------------------------
// Dense GEMV y = A @ x on MI455X (gfx1250), A: 8192x8192 fp32 (256 MB).
// HBM-bound: floor = 256MB / 23.3TB/s ~= 11us. MAC work done with
// V_WMMA_F32_16X16X4_F32 (fp32 in/out, D = A(16x4) * B(4x16) + C).
//
// Layout mapping (CDNA5 ISA, 32-bit A 16x4):
//   lane m   (0..15) : row m, K=0 (vgpr0), K=1 (vgpr1)
//   lane m+16        : row m, K=2 (vgpr0), K=3 (vgpr1)
// -> each lane streams 8 contiguous bytes/step from its own row (16 rows/wave).
// B is x[k..k+3] broadcast across all 16 columns -> every column of D equals
// the partial y vector; lane 0 holds rows 0..7 (acc[0..7]), lane 16 rows 8..15.
//
// Grid: 512 WGs x 256 thr (8 wave32). WG g -> rows [16g,16g+16);
// wave w -> K slice [1024w, 1024w+1024) (256 wmma steps); LDS reduce across
// the 8 waves at the end. x staged once in LDS (32KB), B reads are broadcasts.
// ---------------------------------------------------------------------------

typedef __attribute__((ext_vector_type(2))) float v2f;
typedef __attribute__((ext_vector_type(4))) float v4f;
typedef __attribute__((ext_vector_type(8))) float v8f;

#define NDIM 8192
#define ROWS_PER_WG 16
#define WAVES_PER_WG 8
#define THREADS_PER_WG (WAVES_PER_WG * 32)
#define K_PER_WAVE (NDIM / WAVES_PER_WG) /* 1024 */

__global__ __launch_bounds__(THREADS_PER_WG) void gemv_wmma_f32(
    const float* __restrict__ A, const float* __restrict__ x,
    float* __restrict__ y) {
  __shared__ float xs[NDIM];                            // 32 KB staged x
  __shared__ float partial[WAVES_PER_WG][ROWS_PER_WG];  // cross-wave reduce

  const int tid = threadIdx.x;

  // Stage x into LDS with 16B loads (x is 256B-aligned from hipMalloc).
  {
    const v4f* __restrict__ x4 = (const v4f*)x;
    v4f* xs4 = (v4f*)xs;
#pragma unroll
    for (int i = 0; i < NDIM / 4 / THREADS_PER_WG; ++i)
      xs4[tid + i * THREADS_PER_WG] = x4[tid + i * THREADS_PER_WG];
  }
  __syncthreads();

  const int wave = tid >> 5;
  const int lane = tid & 31;
  const int m = lane & 15;   // row within 16-row tile
  const int h = lane >> 4;   // K half: 0 -> K=0,1 ; 1 -> K=2,3

  const int row0 = blockIdx.x * ROWS_PER_WG;
  const int kbase = wave * K_PER_WAVE;

  // Per-lane streaming pointers. Offsets multiple of 8B -> aligned v2f loads.
  const float* __restrict__ ap =
      A + (size_t)(row0 + m) * NDIM + (size_t)(kbase + 2 * h);
  const float* __restrict__ xp = xs + kbase + 2 * h;

  v8f acc = {};

  // 256 wmma steps per wave; chunk by 4 so we prefetch the A stream ~256B
  // ahead once per chunk (global_prefetch_b8 on gfx1250).
  for (int k = 0; k < K_PER_WAVE; k += 16) {
    __builtin_prefetch(ap + 64, 0, 0);
#pragma unroll
    for (int j = 0; j < 4; ++j) {
      v2f a = *(const v2f*)(ap + 4 * j);        // A[row][k+2h], A[row][k+2h+1]
      v2f b = {xp[4 * j], xp[4 * j + 1]};       // x[k+2h], x[k+2h+1] (bcast)
      // (neg_a, A, neg_b, B, c_mod, C, reuse_a, reuse_b)
      acc = __builtin_amdgcn_wmma_f32_16x16x4_f32(false, a, false, b,
                                                  (short)0, acc, false, false);
    }
    ap += 16;
    xp += 16;
  }

  // All 16 columns of D are identical; column 0 lives in lanes 0 and 16.
  if (lane == 0) {
#pragma unroll
    for (int i = 0; i < 8; ++i) partial[wave][i] = acc[i];       // rows 0..7
  } else if (lane == 16) {
#pragma unroll
    for (int i = 0; i < 8; ++i) partial[wave][8 + i] = acc[i];   // rows 8..15
  }
  __syncthreads();

  // Sum the 8 per-wave partials for each of the 16 rows.
  if (tid < ROWS_PER_WG) {
    float s = 0.f;
#pragma unroll
    for (int w = 0; w < WAVES_PER_WG; ++w) s += partial[w][tid];
    y[row0 + tid] = s;
  }
}

extern "C" void kernel_launch(void* const* d_in, const int* in_sizes, int n_in,
                              void* d_out, int out_size, void* d_ws,
                              size_t ws_size, hipStream_t stream) {
  // setup_inputs order: {"x": [8192] f32, "sparse_matrix": [8192,8192] f32}
  const float* x = (const float*)d_in[0];
  const float* A = (const float*)d_in[1];
  float* y = (float*)d_out;

  dim3 grid(NDIM / ROWS_PER_WG);  // 512 workgroups
  dim3 block(THREADS_PER_WG);     // 256 threads = 8 wave32
  gemv_wmma_f32<<<grid, block, 0, stream>>>(A, x, y);
}